// BatchedSequences_26525718020104
// MI455X (gfx1250) — compile-verified
//
#include <hip/hip_runtime.h>

// ---------------------------------------------------------------------------
// Pad-and-scatter: out[b, p, :] = S[off[b] + p, :] for p < L[b], else 0.
// B = 32, max_sl = 4096, F = 512 floats (128 float4 per row).
// Pure memory movement: ~451 MB total traffic -> ~19 us at 23.3 TB/s (HBM
// roofline; zero FLOPs, so WMMA is irrelevant — the right CDNA5 path is the
// ASYNCcnt-tracked global<->LDS DMA, which moves data without touching VGPRs).
//
// Geometry: each block covers CHUNKS*THREADS float4 = 16 output rows; batch
// boundaries are at row multiples of 4096 and 16 | 4096, so the batch index b
// is BLOCK-UNIFORM -> off[b], off[b+1] become scalar (s_load) reads.
// Each 32-lane wave covers a contiguous 32-aligned float4 range that never
// crosses a 128-f4 row, so the copy-vs-zero decision is wave-uniform and
// EXEC is all-ones around the async instructions.
// ---------------------------------------------------------------------------

typedef float vfloat4 __attribute__((ext_vector_type(4)));

constexpr int MAX_SL     = 4096;   // rows per batch in output
constexpr int F4_PER_ROW = 128;    // 512 floats / 4
constexpr int THREADS    = 256;    // 8 waves (wave32)
constexpr int CHUNKS     = 8;      // float4 chunks per thread (8 outstanding
                                   // async loads per lane before the wait)
constexpr int ROWS_PER_BLOCK = THREADS * CHUNKS / F4_PER_ROW;  // 16

// Tiny prologue: exclusive prefix sum of L into off[0..B] (B = 32).
__global__ void offsets_kernel(const int* __restrict__ L,
                               int* __restrict__ off, int B) {
  if (blockIdx.x == 0 && threadIdx.x == 0) {
    int acc = 0;
    for (int i = 0; i < B; ++i) { off[i] = acc; acc += L[i]; }
    off[B] = acc;
  }
}

__global__ __launch_bounds__(THREADS) void pad_scatter_kernel(
    const float* __restrict__ S, const int* __restrict__ off,
    float* __restrict__ out) {
  // 16 bytes of private LDS staging per (thread, chunk): 32 KB per block.
  __shared__ alignas(16) unsigned char stage[THREADS * CHUNKS * 16];

  const unsigned t = threadIdx.x;
  const unsigned long long base_f4 =
      (unsigned long long)blockIdx.x * (THREADS * CHUNKS);
  const unsigned long long s_base   = (unsigned long long)(size_t)S;
  const unsigned long long out_base = (unsigned long long)(size_t)out;

  // Block-uniform batch index -> scalar loads of the offset table.
  const unsigned base_row = blockIdx.x * ROWS_PER_BLOCK;
  const unsigned b        = base_row >> 12;               // / MAX_SL (uniform)
  const int o0  = off[b];                                 // s_load
  const int len = off[b + 1] - o0;                        // SGPR

  bool valid[CHUNKS];

  // Phase 1: async 128-bit DMA loads HBM -> LDS for valid chunks; streaming
  // nontemporal zero stores for padding chunks.
#pragma unroll
  for (int i = 0; i < CHUNKS; ++i) {
    const unsigned long long f4 = base_f4 + (unsigned long long)i * THREADS + t;
    const unsigned long long row = f4 >> 7;               // / F4_PER_ROW
    const unsigned col = (unsigned)(f4 & (F4_PER_ROW - 1));
    const unsigned p   = (unsigned)(row & (MAX_SL - 1));
    valid[i] = ((int)p < len);

    if (valid[i]) {
      const unsigned long long srow = (unsigned long long)(o0 + (int)p);
      const unsigned long long src =
          s_base + (srow * F4_PER_ROW + col) * 16ull;
      const unsigned lds_addr =
          (unsigned)(size_t)(&stage[((unsigned)i * THREADS + t) * 16]);
      asm volatile("global_load_async_to_lds_b128 %0, %1, off"
                   :
                   : "v"(lds_addr), "v"(src)
                   : "memory");
    } else {
      vfloat4 z = {0.0f, 0.0f, 0.0f, 0.0f};
      __builtin_nontemporal_store(z, (vfloat4*)(out + f4 * 4));
    }
  }

  // Fence this wave's async loads; LDS slots are thread-private, so a
  // per-wave counter wait suffices (no workgroup barrier needed).
  asm volatile("s_wait_asynccnt 0x0" ::: "memory");

  // Phase 2: async-store the staged tiles LDS -> HBM. S_ENDPGM performs an
  // implicit wait-idle, draining ASYNCcnt before the wave retires.
#pragma unroll
  for (int i = 0; i < CHUNKS; ++i) {
    if (valid[i]) {
      const unsigned long long f4 =
          base_f4 + (unsigned long long)i * THREADS + t;
      const unsigned long long dst = out_base + f4 * 16ull;
      const unsigned lds_addr =
          (unsigned)(size_t)(&stage[((unsigned)i * THREADS + t) * 16]);
      asm volatile("global_store_async_from_lds_b128 %0, %1, off"
                   :
                   : "v"(dst), "v"(lds_addr)
                   : "memory");
    }
  }
}

extern "C" void kernel_launch(void* const* d_in, const int* in_sizes, int n_in,
                              void* d_out, int out_size, void* d_ws,
                              size_t ws_size, hipStream_t stream) {
  const float* S = (const float*)d_in[0];
  const int* L   = (const int*)d_in[1];   // (B,1) int32 sequence lengths
  float* out     = (float*)d_out;
  int* off       = (int*)d_ws;            // needs (B+1)*4 = 132 bytes

  const int B = in_sizes[1];              // 32

  offsets_kernel<<<1, 1, 0, stream>>>(L, off, B);

  // out_size = B * MAX_SL * 512 floats; each block moves THREADS*CHUNKS
  // float4s: 16,777,216 f4 / 2048 = 8192 blocks (exact cover).
  const long long total_f4 = (long long)out_size / 4;
  const int blocks = (int)(total_f4 / ((long long)THREADS * CHUNKS));
  pad_scatter_kernel<<<blocks, THREADS, 0, stream>>>(S, off, out);
}